// TorchLIFNeuronGroup_85152021610615
// MI455X (gfx1250) — compile-verified
//
#include <hip/hip_runtime.h>
#include <cstdint>

// LIF neuron scan: 2048 sequential rows over 16384 independent neuron chains.
// Memory-bound streaming (~402 MB @ 23.3 TB/s ~= 17 us). Parallelism is capped
// at 512 wave32s, so we (a) hide HBM latency with the CDNA5 async-to-LDS path
// (ASYNCcnt, 60 ops in flight/wave) + long-range L2 prefetch, and (b) minimize
// per-row instructions via an unrolled ring with immediate DS/store offsets.

#define N_NEURONS   16384
#define BATCH       2048
#define BLOCK       64            // 2 wave32s per block, 256 blocks
#define STAGES      32            // LDS ring slots (power of 2)
#define PD          30            // rows in flight: 2 async/row -> 60 <= ASYNCcnt max 63
#define PF          96            // L2 prefetch distance in rows
#define ROW_BYTES   (N_NEURONS * 4)
#define STAGE_BYTES (BLOCK * 8)   // float2 per thread per stage

__global__ __launch_bounds__(BLOCK) void lif_scan_kernel(
    const float* __restrict__ in_cur,
    const float* __restrict__ noise,
    float* __restrict__ out_spk)
{
    __shared__ float2 ring[STAGES][BLOCK];   // .x = input, .y = noise (16 KB)

    const int tid = threadIdx.x;
    const int j   = blockIdx.x * BLOCK + tid;           // neuron index
    const unsigned g0 = (unsigned)j * 4u;               // byte offset within a row
    const unsigned l0 = (unsigned)(uintptr_t)(&ring[0][tid]);  // LDS byte offset

    // Issue the two async copies (input -> .x, noise -> .y) for row r, stage s.
    auto issue = [&](int r, int s) {
        const unsigned goff = (unsigned)r * (unsigned)ROW_BYTES + g0;
        const unsigned li   = l0 + (unsigned)s * (unsigned)STAGE_BYTES;
        asm volatile("global_load_async_to_lds_b32 %0, %1, %2"
                     :: "v"(li), "v"(goff), "s"(in_cur) : "memory");
        asm volatile("global_load_async_to_lds_b32 %0, %1, %2"
                     :: "v"(li + 4u), "v"(goff), "s"(noise) : "memory");
    };

    float V   = 0.0f;                 // V_RESET
    float Vth = 1.0f;                 // V_TH0
    const float K = 1.0f / 20.0f;     // DT / TAU

    // Consume ring stage s (= row t & 31), advance state, emit spike.
    auto step = [&](int t, int s) {
        const float2 d = ring[s][tid];                  // one ds_load_b64
        V = __builtin_fmaf(K, d.x - V, V) + d.y;        // leak + input + noise
        const bool  sp    = (V >= Vth);
        const float spike = sp ? 1.0f : 0.0f;
        V   = sp ? 0.0f : V;                            // reset on spike
        // lower clip at 0.5 is dead code: Vth starts at 1.0 and never decreases
        Vth = fminf(__builtin_fmaf(0.1f, spike, Vth), 2.0f);
        __builtin_nontemporal_store(spike, out_spk + (size_t)t * N_NEURONS + (size_t)j);
    };

    // One 32-row chunk: ring stage == u is a compile-time constant, so DS reads
    // and NT stores use immediate offsets; only async issues need per-row adds.
    auto chunk = [&](int tbase, bool pf) {
#pragma unroll
        for (int u = 0; u < STAGES; ++u) {
            const int t = tbase + u;
            if (pf) {
                __builtin_prefetch(in_cur + (size_t)(t + PF) * N_NEURONS + (size_t)j, 0, 1);
                __builtin_prefetch(noise  + (size_t)(t + PF) * N_NEURONS + (size_t)j, 0, 1);
            }
            asm volatile("s_wait_asynccnt 58" ::: "memory");   // oldest row arrived
            step(t, u);
            issue(t + PD, (u + PD) & (STAGES - 1));            // slot freed 2 rows ago
        }
    };

    // ---- prologue: fill the pipeline (rows 0..PD-1, 2*PD = 60 outstanding) ----
#pragma unroll
    for (int s = 0; s < PD; ++s)
        issue(s, s);

    // ---- steady state: rows [0, 1984) in 32-row chunks ----
    const int CHUNKS    = (BATCH / STAGES) - 2;      // 62 chunks
    const int PF_CHUNKS = (BATCH - PF) / STAGES;     // 61: prefetch stays in bounds
    for (int c = 0; c < PF_CHUNKS; ++c) chunk(c * STAGES, true);
    for (int c = PF_CHUNKS; c < CHUNKS; ++c) chunk(c * STAGES, false);

    // ---- transition: rows [1984, 2018) still issue rows up to 2047 ----
#pragma unroll 1
    for (int t = CHUNKS * STAGES; t < BATCH - PD; ++t) {
        asm volatile("s_wait_asynccnt 58" ::: "memory");
        step(t, t & (STAGES - 1));
        issue(t + PD, (t + PD) & (STAGES - 1));
    }

    // ---- drain: everything left was issued >= PD rows ago ----
    asm volatile("s_wait_asynccnt 0" ::: "memory");
#pragma unroll 1
    for (int t = BATCH - PD; t < BATCH; ++t)
        step(t, t & (STAGES - 1));
}

extern "C" void kernel_launch(void* const* d_in, const int* in_sizes, int n_in,
                              void* d_out, int out_size, void* d_ws, size_t ws_size,
                              hipStream_t stream) {
    const float* in_cur = (const float*)d_in[0];   // [2048, 16384] f32
    const float* noise  = (const float*)d_in[1];   // [2048, 16384] f32
    float*       out    = (float*)d_out;           // [2048, 16384] f32 spikes

    dim3 grid(N_NEURONS / BLOCK);                  // 256 blocks
    dim3 block(BLOCK);                             // 64 threads = 2 wave32s
    lif_scan_kernel<<<grid, block, 0, stream>>>(in_cur, noise, out);
}